// DemandExtraction_37056977829943
// MI455X (gfx1250) — compile-verified
//
#include <hip/hip_runtime.h>
#include <hip/hip_bf16.h>
#include <math.h>

// ---------------------------------------------------------------------------
// DemandExtraction forward for MI455X (gfx1250, wave32).
// GEMMs: V_WMMA_F32_16X16X4_F32, 16x64 tile per wave, branch-free inner loop,
// software-pipelined (next k-step fragments prefetched across current WMMAs).
// ---------------------------------------------------------------------------

typedef float v2f __attribute__((ext_vector_type(2)));
typedef float v8f __attribute__((ext_vector_type(8)));

#define B_DIM 32
#define S_DIM 50
#define D_DIM 4
#define H_DIM 128
#define E_DIM 128
#define I_DIM 5000

// -------------------------- embedding gather -------------------------------
__global__ void gather_rows_kernel(float* __restrict__ dst,
                                   const float* __restrict__ table,
                                   const int* __restrict__ idx,
                                   int n_rows) {
  int t = blockIdx.x * blockDim.x + threadIdx.x;
  if (t >= n_rows * E_DIM) return;
  int r = t >> 7;      // /128
  int c = t & 127;
  dst[t] = table[(long)idx[r] * E_DIM + c];
}

// -------------------- generic C = A(Mx128) * W^T (+bias) -------------------
// W row-major (N_total x ldw): W[n][k] = W[n*ldw + k]. C row-major (M x N).
// One 16(M) x 64(N) tile per wave: 1 A-frag feeds 4 WMMAs per k-step.
__global__ void gemm_awt_wmma_kernel(float* __restrict__ C,
                                     const float* __restrict__ A,
                                     const float* __restrict__ W,
                                     const float* __restrict__ bias,
                                     int M, int N, int ldw) {
  const int K = 128;
  int wave = blockIdx.x * (blockDim.x >> 5) + (threadIdx.x >> 5);
  int lane = threadIdx.x & 31;
  int tilesN = N >> 6;              // 64-wide N tiles (N is a multiple of 64)
  int tilesM = (M + 15) >> 4;
  if (wave >= tilesM * tilesN) return;   // wave-uniform: EXEC stays all-1s
  int tm = wave / tilesN;
  int tn = wave - tm * tilesN;
  int m0 = tm << 4;
  int n0 = tn << 6;

  int half = lane >> 4;             // 0 -> K+0..1, 1 -> K+2..3
  int l15  = lane & 15;

  // Row-clamped A pointer: OOB rows compute garbage that is never stored.
  int arow = m0 + l15;
  if (arow > M - 1) arow = M - 1;
  const v2f* ap  = (const v2f*)(A + (long)arow * K);
  const v2f* wp0 = (const v2f*)(W + (long)(n0 +  0 + l15) * ldw);
  const v2f* wp1 = (const v2f*)(W + (long)(n0 + 16 + l15) * ldw);
  const v2f* wp2 = (const v2f*)(W + (long)(n0 + 32 + l15) * ldw);
  const v2f* wp3 = (const v2f*)(W + (long)(n0 + 48 + l15) * ldw);

  v8f acc0 = {0.f,0.f,0.f,0.f,0.f,0.f,0.f,0.f};
  v8f acc1 = acc0, acc2 = acc0, acc3 = acc0;

  // ---- software pipeline: prologue loads k=0 fragments ----
  v2f a  = ap[half];
  v2f b0 = wp0[half];
  v2f b1 = wp1[half];
  v2f b2 = wp2[half];
  v2f b3 = wp3[half];

#pragma unroll 4
  for (int k = 4; k < K; k += 4) {
    int kv = (k >> 1) + half;       // float2 index of K+2*half
    // prefetch next k-step before consuming current fragments
    v2f an  = ap[kv];
    v2f bn0 = wp0[kv];
    v2f bn1 = wp1[kv];
    v2f bn2 = wp2[kv];
    v2f bn3 = wp3[kv];
    acc0 = __builtin_amdgcn_wmma_f32_16x16x4_f32(false, a, false, b0, (short)0, acc0, false, false);
    acc1 = __builtin_amdgcn_wmma_f32_16x16x4_f32(false, a, false, b1, (short)0, acc1, false, false);
    acc2 = __builtin_amdgcn_wmma_f32_16x16x4_f32(false, a, false, b2, (short)0, acc2, false, false);
    acc3 = __builtin_amdgcn_wmma_f32_16x16x4_f32(false, a, false, b3, (short)0, acc3, false, false);
    a = an; b0 = bn0; b1 = bn1; b2 = bn2; b3 = bn3;
  }
  // ---- epilogue: last k-step ----
  acc0 = __builtin_amdgcn_wmma_f32_16x16x4_f32(false, a, false, b0, (short)0, acc0, false, false);
  acc1 = __builtin_amdgcn_wmma_f32_16x16x4_f32(false, a, false, b1, (short)0, acc1, false, false);
  acc2 = __builtin_amdgcn_wmma_f32_16x16x4_f32(false, a, false, b2, (short)0, acc2, false, false);
  acc3 = __builtin_amdgcn_wmma_f32_16x16x4_f32(false, a, false, b3, (short)0, acc3, false, false);

  int c0 = n0 +  0 + l15;
  int c1 = n0 + 16 + l15;
  int c2 = n0 + 32 + l15;
  int c3 = n0 + 48 + l15;
  float bv0 = bias ? bias[c0] : 0.f;
  float bv1 = bias ? bias[c1] : 0.f;
  float bv2 = bias ? bias[c2] : 0.f;
  float bv3 = bias ? bias[c3] : 0.f;

  int rbase = m0 + (half << 3);     // lanes 16-31 hold rows r+8
  if (m0 + 16 <= M) {               // full tile: unguarded stores
#pragma unroll
    for (int r = 0; r < 8; ++r) {
      long ro = (long)(rbase + r) * N;
      C[ro + c0] = acc0[r] + bv0;
      C[ro + c1] = acc1[r] + bv1;
      C[ro + c2] = acc2[r] + bv2;
      C[ro + c3] = acc3[r] + bv3;
    }
  } else {                          // boundary tile only
#pragma unroll
    for (int r = 0; r < 8; ++r) {
      int row = rbase + r;
      if (row < M) {
        long ro = (long)row * N;
        C[ro + c0] = acc0[r] + bv0;
        C[ro + c1] = acc1[r] + bv1;
        C[ro + c2] = acc2[r] + bv2;
        C[ro + c3] = acc3[r] + bv3;
      }
    }
  }
}

// ------------------- agg = log(sum_s exp(hidden)) (B,512) ------------------
__global__ void agg_lse_kernel(float* __restrict__ agg,
                               const float* __restrict__ hidden) {
  int t = blockIdx.x * blockDim.x + threadIdx.x;
  if (t >= B_DIM * 512) return;
  int b = t >> 9;
  int n = t & 511;
  const float* hp = hidden + (long)b * S_DIM * 512 + n;
  float s = 0.f;
#pragma unroll 5
  for (int i = 0; i < S_DIM; ++i) s += expf(hp[(long)i * 512]);
  agg[t] = logf(s);
}

// --------------- per-batch off-diagonal Gram contribution ------------------
__global__ void simloss_partial_kernel(float* __restrict__ partial,
                                       const float* __restrict__ agg) {
  __shared__ float sagg[512];
  __shared__ float red[128];
  __shared__ float nrm[D_DIM];
  int b = blockIdx.x;
  int tid = threadIdx.x;   // 128 threads
  for (int i = tid; i < 512; i += 128) sagg[i] = agg[b * 512 + i];
  __syncthreads();

  for (int d = 0; d < D_DIM; ++d) {
    float v = sagg[d * 128 + tid];
    red[tid] = v * v;
    __syncthreads();
    for (int s = 64; s > 0; s >>= 1) {
      if (tid < s) red[tid] += red[tid + s];
      __syncthreads();
    }
    if (tid == 0) nrm[d] = fmaxf(sqrtf(red[0]), 1e-8f);
    __syncthreads();
  }

  float total = 0.f;   // meaningful on tid 0 only
  for (int d = 0; d < D_DIM; ++d) {
    for (int e = d + 1; e < D_DIM; ++e) {
      red[tid] = sagg[d * 128 + tid] * sagg[e * 128 + tid];
      __syncthreads();
      for (int s = 64; s > 0; s >>= 1) {
        if (tid < s) red[tid] += red[tid + s];
        __syncthreads();
      }
      if (tid == 0) total += 2.f * red[0] / (nrm[d] * nrm[e]);
      __syncthreads();
    }
  }
  if (tid == 0) partial[b] = total;
}

__global__ void simloss_final_kernel(float* __restrict__ out_loss,
                                     const float* __restrict__ partial) {
  if (threadIdx.x == 0 && blockIdx.x == 0) {
    float s = 0.f;
    for (int b = 0; b < B_DIM; ++b) s += partial[b];   // deterministic order
    *out_loss = s / (float)(B_DIM * D_DIM * (D_DIM - 1));
  }
}

// ------------- demand_score[b,d,s] : one wave32 per output -----------------
__global__ void score_kernel(float* __restrict__ out,            // (B,D,S)
                             const float* __restrict__ aggp,     // (B*D,128)
                             const float* __restrict__ keyp,     // (B*S*D,128)
                             const float* __restrict__ wsc) {
  int wave = blockIdx.x * (blockDim.x >> 5) + (threadIdx.x >> 5);
  int lane = threadIdx.x & 31;
  if (wave >= B_DIM * D_DIM * S_DIM) return;
  int s  = wave % S_DIM;
  int bd = wave / S_DIM;          // b*4 + d
  int b  = bd >> 2;
  int d  = bd & 3;
  const float4* ap = (const float4*)(aggp + bd * 128);
  const float4* kp = (const float4*)(keyp + ((long)((b * S_DIM + s) * D_DIM + d) << 7));
  const float4* wp = (const float4*)wsc;
  float4 av = ap[lane];
  float4 kv = kp[lane];
  float4 wv = wp[lane];
  float acc = fmaxf(av.x + kv.x, 0.f) * wv.x
            + fmaxf(av.y + kv.y, 0.f) * wv.y
            + fmaxf(av.z + kv.z, 0.f) * wv.z
            + fmaxf(av.w + kv.w, 0.f) * wv.w;
  for (int m = 16; m > 0; m >>= 1) acc += __shfl_xor(acc, m, 32);
  if (lane == 0) out[wave] = acc;   // wave == bd*50 + s == flat (b,d,s)
}

// --------- demand_score_candidate[b,d,i] : one wave32 per output -----------
__global__ void cand_score_kernel(float* __restrict__ out,          // (B,D,I)
                                  const float* __restrict__ aggp,   // (B*D,128)
                                  const float* __restrict__ candp,  // (I*D,128)
                                  const float* __restrict__ wsc) {
  int wave = blockIdx.x * (blockDim.x >> 5) + (threadIdx.x >> 5);
  int lane = threadIdx.x & 31;
  if (wave >= B_DIM * D_DIM * I_DIM) return;
  int i  = wave % I_DIM;
  int bd = wave / I_DIM;
  int d  = bd & 3;
  const float4* ap = (const float4*)(aggp + bd * 128);
  const float4* cp = (const float4*)(candp + ((long)(i * D_DIM + d) << 7));
  const float4* wp = (const float4*)wsc;
  float4 av = ap[lane];
  float4 cv = cp[lane];
  float4 wv = wp[lane];
  float acc = fmaxf(av.x + cv.x, 0.f) * wv.x
            + fmaxf(av.y + cv.y, 0.f) * wv.y
            + fmaxf(av.z + cv.z, 0.f) * wv.z
            + fmaxf(av.w + cv.w, 0.f) * wv.w;
  for (int m = 16; m > 0; m >>= 1) acc += __shfl_xor(acc, m, 32);
  if (lane == 0) out[wave] = acc;   // wave == bd*5000 + i == flat (b,d,i)
}

// ---------------------------------------------------------------------------
extern "C" void kernel_launch(void* const* d_in, const int* in_sizes, int n_in,
                              void* d_out, int out_size, void* d_ws, size_t ws_size,
                              hipStream_t stream) {
  (void)in_sizes; (void)n_in; (void)out_size; (void)ws_size;
  const int*   inp   = (const int*)d_in[0];     // (32,50)
  const int*   cand  = (const int*)d_in[1];     // (5000,)
  const float* table = (const float*)d_in[2];   // (10000,128)
  const float* Wdem  = (const float*)d_in[3];   // (512,128)
  const float* Wsc   = (const float*)d_in[4];   // (128,256) [Wd | Wk]
  const float* bsc   = (const float*)d_in[5];   // (128,)
  const float* wsc   = (const float*)d_in[6];   // (128,)

  float* out      = (float*)d_out;
  float* o_score  = out;                 //   6400  (32,4,50)
  float* o_cscore = out + 6400;          // 640000  (32,4,5000)
  float* o_emb    = out + 646400;        // 204800  (32,50,128)
  float* o_cemb   = out + 851200;        // 640000  (5000,128)
  float* o_loss   = out + 1491200;       //      1

  float* ws      = (float*)d_ws;
  float* hidden  = ws;                   //  819200  (1600,512) == (6400,128)
  float* chid    = hidden + 819200;      // 2560000  (5000,512) == (20000,128)
  float* agg     = chid + 2560000;       //   16384  (32,512)   == (128,128)
  float* aggp    = agg + 16384;          //   16384  (128,128)
  float* keyp    = aggp + 16384;         //  819200  (6400,128)
  float* candp   = keyp + 819200;        // 2560000  (20000,128)
  float* partial = candp + 2560000;      //      32

  // 1) gathers (emb / cand_emb are also outputs)
  gather_rows_kernel<<<(1600 * 128 + 255) / 256, 256, 0, stream>>>(o_emb, table, inp, 1600);
  gather_rows_kernel<<<(5000 * 128 + 255) / 256, 256, 0, stream>>>(o_cemb, table, cand, 5000);

  // 2) hidden GEMMs: X @ W_demand^T   (N=512, ldw=128)
  {
    int waves = ((1600 + 15) / 16) * (512 / 64);                 // 800
    gemm_awt_wmma_kernel<<<(waves + 7) / 8, 256, 0, stream>>>(
        hidden, o_emb, Wdem, nullptr, 1600, 512, 128);
  }
  {
    int waves = ((5000 + 15) / 16) * (512 / 64);                 // 2504
    gemm_awt_wmma_kernel<<<(waves + 7) / 8, 256, 0, stream>>>(
        chid, o_cemb, Wdem, nullptr, 5000, 512, 128);
  }

  // 3) agg = log(sum_s exp(hidden))
  agg_lse_kernel<<<(B_DIM * 512 + 255) / 256, 256, 0, stream>>>(agg, hidden);

  // 4) projections (W_score: Wd = cols [0,128), Wk = cols [128,256), ldw=256)
  {
    int waves = (128 / 16) * (128 / 64);                         // 16
    gemm_awt_wmma_kernel<<<(waves + 7) / 8, 256, 0, stream>>>(
        aggp, agg, Wsc, bsc, 128, 128, 256);                     // agg_proj + b_score
  }
  {
    int waves = (6400 / 16) * (128 / 64);                        // 800
    gemm_awt_wmma_kernel<<<(waves + 7) / 8, 256, 0, stream>>>(
        keyp, hidden, Wsc + 128, nullptr, 6400, 128, 256);       // key_proj (Wk)
  }
  {
    int waves = (20000 / 16) * (128 / 64);                       // 2500
    gemm_awt_wmma_kernel<<<(waves + 7) / 8, 256, 0, stream>>>(
        candp, chid, Wsc + 128, nullptr, 20000, 128, 256);       // cand_proj (Wk)
  }

  // 5) demand similarity loss
  simloss_partial_kernel<<<B_DIM, 128, 0, stream>>>(partial, agg);
  simloss_final_kernel<<<1, 32, 0, stream>>>(o_loss, partial);

  // 6) scores
  {
    int waves = B_DIM * D_DIM * S_DIM;                           // 6400
    score_kernel<<<(waves + 7) / 8, 256, 0, stream>>>(o_score, aggp, keyp, wsc);
  }
  {
    int waves = B_DIM * D_DIM * I_DIM;                           // 640000
    cand_score_kernel<<<(waves + 7) / 8, 256, 0, stream>>>(o_cscore, aggp, candp, wsc);
  }
}